// MessagePassing_8735963480380
// MI455X (gfx1250) — compile-verified
//
#include <hip/hip_runtime.h>
#include <hip/hip_bf16.h>
#include <stdint.h>

// MessagePassing scatter-add:  out[dst[e]] += x[src[e]]  for e in [0,E), D=64 fp32.
//
// MI455X plan:
//  - x (25.6 MB) and out (25.6 MB) are L2-resident (192 MB L2); the kernel is
//    bound by L2 gather + atomic-RMW throughput, not HBM (~70 MB compulsory).
//  - Feature work: 16 lanes/edge x float4/lane -> one global_load_b128 gather
//    and 4 non-returning global_atomic_add_f32 per lane. Wave32: one wave
//    covers 2 edges.
//  - Index stream (19.2 MB, single-use) is staged into LDS by the Tensor Data
//    Mover: one D# describes a 2 x CHUNK tile of the (2,E) int32 edge_index
//    (tile_dim1=2, tensor_dim0_stride=E), double-buffered and tracked with
//    TENSORcnt, so the index-fetch latency is off the critical path.

typedef uint32_t u32;
typedef __attribute__((ext_vector_type(4))) u32 u32x4;
typedef __attribute__((ext_vector_type(4))) int i32x4;
typedef __attribute__((ext_vector_type(8))) int i32x8;

#define D_FEAT          64
#define THREADS         256
#define LANES_PER_EDGE  16
#define EDGES_PER_STEP  (THREADS / LANES_PER_EDGE)   // 16 edges per block per inner step
#define CHUNK           1024                          // edges staged per TDM descriptor

// Issue one TDM load: 2 rows (src,dst) x CHUNK int32 from the (2,E) edge_index
// tensor into LDS at lds_byte_addr (row 0 -> [0,CHUNK), row 1 -> [CHUNK,2*CHUNK)).
__device__ __forceinline__ void tdm_load_idx_chunk(const int* gsrc_chunk,
                                                   u32 remaining_in_row,
                                                   u32 row_stride_elems,
                                                   u32 lds_byte_addr) {
    uint64_t ga = (uint64_t)(uintptr_t)gsrc_chunk;

    // D# group 0: count=1 | lds_addr | global_addr[56:0] | type=2
    u32x4 g0;
    g0.x = 1u;                                            // count=1, user mode
    g0.y = lds_byte_addr;                                 // bits [63:32]
    g0.z = (u32)ga;                                       // global_addr[31:0]
    g0.w = ((u32)(ga >> 32) & 0x01FFFFFFu) | 0x80000000u; // ga[56:32] | type=2<<30

    // D# group 1
    i32x8 g1;
    g1[0] = (int)(2u << 16);                              // data_size=2 (4 bytes)
    g1[1] = (int)((remaining_in_row & 0xFFFFu) << 16);    // tensor_dim0[15:0] @ [63:48]
    g1[2] = (int)((remaining_in_row >> 16) | (2u << 16)); // tensor_dim0[31:16] | tensor_dim1=2
    g1[3] = (int)((u32)CHUNK << 16);                      // tile_dim0 @ [127:112]
    g1[4] = 2;                                            // tile_dim1=2 @ [143:128]
    g1[5] = (int)row_stride_elems;                        // tensor_dim0_stride[31:0]
    g1[6] = 0;
    g1[7] = 0;

    i32x4 z4 = {0, 0, 0, 0};
#if __clang_major__ >= 23
    i32x8 z8 = {0, 0, 0, 0, 0, 0, 0, 0};
    __builtin_amdgcn_tensor_load_to_lds(g0, g1, z4, z4, z8, 0);
#else
    __builtin_amdgcn_tensor_load_to_lds(g0, g1, z4, z4, 0);
#endif
}

__global__ __launch_bounds__(THREADS)
void MessagePassing_zero_kernel(float4* __restrict__ out4, int n4) {
    int i = blockIdx.x * blockDim.x + threadIdx.x;
    if (i < n4) {
        out4[i] = make_float4(0.f, 0.f, 0.f, 0.f);
    }
}

__global__ __launch_bounds__(THREADS)
void MessagePassing_scatter_kernel(const float* __restrict__ x,
                                   const int*   __restrict__ src,   // edge_index row 0
                                   float*       __restrict__ out,
                                   int n_edges, int n_chunks) {
    __shared__ int lds_idx[2][2 * CHUNK];   // [buffer][src row | dst row]

    const int lane_c = (threadIdx.x & (LANES_PER_EDGE - 1)) << 2;  // feature offset 0..60
    const int sub    = threadIdx.x >> 4;                           // edge slot 0..15
    const bool issuer = (threadIdx.x < 32);                        // wave 0 drives the TDM

    int chunk = blockIdx.x;

    // Prologue: stage the first chunk.
    if (issuer) {
        tdm_load_idx_chunk(src + (size_t)chunk * CHUNK,
                           (u32)(n_edges - chunk * CHUNK),
                           (u32)n_edges,
                           (u32)(uintptr_t)&lds_idx[0][0]);
    }

    int it = 0;
    for (; chunk < n_chunks; chunk += gridDim.x, ++it) {
        const int next = chunk + (int)gridDim.x;
        if (issuer) {
            if (next < n_chunks) {
                // Stage next chunk into the other buffer, then wait for the
                // current one (TDM ops complete in order -> tensorcnt<=1).
                tdm_load_idx_chunk(src + (size_t)next * CHUNK,
                                   (u32)(n_edges - next * CHUNK),
                                   (u32)n_edges,
                                   (u32)(uintptr_t)&lds_idx[(it + 1) & 1][0]);
                __builtin_amdgcn_s_wait_tensorcnt(1);
            } else {
                __builtin_amdgcn_s_wait_tensorcnt(0);
            }
        }
        __syncthreads();   // publish staged indices to all 8 waves

        const int* bsrc = &lds_idx[it & 1][0];
        const int* bdst = &lds_idx[it & 1][CHUNK];
        const int  base = chunk * CHUNK;
        const int  lim  = min(CHUNK, n_edges - base);

        #pragma unroll 4
        for (int i = sub; i < lim; i += EDGES_PER_STEP) {
            const int s = bsrc[i];   // same-address LDS read broadcasts across 16 lanes
            const int d = bdst[i];

            // 16 B gather of the source row (256 B-aligned row base + 16 B-aligned lane_c*4).
            const float4 v = *reinterpret_cast<const float4*>(x + (size_t)s * D_FEAT + lane_c);

            // Non-returning device-scope f32 atomic adds into the target row (L2 RMW).
            float* o = out + (size_t)d * D_FEAT + lane_c;
            atomicAdd(o + 0, v.x);
            atomicAdd(o + 1, v.y);
            atomicAdd(o + 2, v.z);
            atomicAdd(o + 3, v.w);
        }
        __syncthreads();   // buffer it&1 may be overwritten next iteration
    }
}

extern "C" void kernel_launch(void* const* d_in, const int* in_sizes, int n_in,
                              void* d_out, int out_size, void* d_ws, size_t ws_size,
                              hipStream_t stream) {
    const float* x  = (const float*)d_in[0];   // [N, 64] fp32
    const int*   ei = (const int*)d_in[1];     // [2, E] flat: row 0 = src, row 1 = dst

    const int n_edges = in_sizes[1] / 2;
    float* out = (float*)d_out;                // [N, 64] fp32

    // 1) Zero the output (harness poisons d_out; scatter-add needs a zero base).
    const int n4 = out_size / 4;
    const int zblocks = (n4 + THREADS - 1) / THREADS;
    MessagePassing_zero_kernel<<<zblocks, THREADS, 0, stream>>>((float4*)out, n4);

    // 2) Scatter-add with TDM-staged, double-buffered index chunks.
    //    >=2 chunks per block so the double buffer actually pipelines.
    const int n_chunks = (n_edges + CHUNK - 1) / CHUNK;
    int blocks = (n_chunks + 1) / 2;
    if (blocks < 1) blocks = 1;
    if (blocks > n_chunks) blocks = n_chunks;
    MessagePassing_scatter_kernel<<<blocks, THREADS, 0, stream>>>(x, ei, out, n_edges, n_chunks);
}